// SmileNet_35871566856845
// MI455X (gfx1250) — compile-verified
//
#include <hip/hip_runtime.h>

typedef float v2f __attribute__((ext_vector_type(2)));
typedef float v8f __attribute__((ext_vector_type(8)));

// gcc-style vector type matching the async-LDS builtin's expected parameter
typedef int v4i_b __attribute__((vector_size(16)));
typedef __attribute__((address_space(1))) v4i_b* as1_v4i;   // global
typedef __attribute__((address_space(3))) v4i_b* as3_v4i;   // LDS

#define B_ROWS 16384
#define D_IN   128
#define D_H    256
#define CDIM   512
#define NSEG   128
#define CONDL  11
#define KFREE  67

// Workspace layout (floats):
//   Wcs  [32*64*32*2 = 131072]  swizzled Wc = W2 @ basis^T fragments
//   W1s  [16*32*32*2 =  32768]  swizzled W1 fragments
//   cbias[512]                  b2 @ basis^T + xstar
#define WCS_FLOATS  131072
#define W1S_FLOATS  32768

#if __has_builtin(__builtin_amdgcn_global_load_async_to_lds_b128)
#define HAVE_ASYNC_LDS 1
#else
#define HAVE_ASYNC_LDS 0
#endif

// ---------------------------------------------------------------------------
// Kernel A: swizzle W1 (128 x 256, row-major) into per-fragment pair layout:
//   W1s[((nt*32 + k)*32 + lane)*2 + {0,1}] = W1[kr + {0,1}][nt*16 + n]
//   with kr = 4k + 2*(lane>>4), n = lane & 15
// ---------------------------------------------------------------------------
__global__ __launch_bounds__(256) void swizzle_w1(
    const float* __restrict__ W1, float* __restrict__ W1s)
{
    int idx = blockIdx.x * blockDim.x + threadIdx.x;   // 0 .. 16*32*32-1
    if (idx >= 16 * 32 * 32) return;
    int nt = idx >> 10;
    int k  = (idx >> 5) & 31;
    int l  = idx & 31;
    int n  = l & 15;
    int kr = 4 * k + ((l >> 4) << 1);
    W1s[idx * 2 + 0] = W1[kr * D_H + nt * 16 + n];
    W1s[idx * 2 + 1] = W1[(kr + 1) * D_H + nt * 16 + n];
}

// ---------------------------------------------------------------------------
// Kernel B: compute Wc = W2 @ basis^T directly into swizzled fragment layout
//   Wcs[((ct*64 + k)*32 + lane)*2 + {0,1}] = Wc[kr + {0,1}][ct*16 + n]
// ---------------------------------------------------------------------------
__global__ __launch_bounds__(256) void wc_swizzled(
    const float* __restrict__ W2, const float* __restrict__ basis,
    float* __restrict__ Wcs)
{
    int idx = blockIdx.x * blockDim.x + threadIdx.x;   // 0 .. 32*64*32-1
    if (idx >= 32 * 64 * 32) return;
    int ct = idx >> 11;
    int k  = (idx >> 5) & 63;
    int l  = idx & 31;
    int n  = l & 15;
    int kr = 4 * k + ((l >> 4) << 1);
    int c  = ct * 16 + n;
    const float* bs = basis + c * KFREE;          // basis (512, 67) row-major
    const float* w0 = W2 + kr * KFREE;            // W2    (256, 67) row-major
    const float* w1 = w0 + KFREE;
    float s0 = 0.f, s1 = 0.f;
    for (int j = 0; j < KFREE; ++j) {
        float b = bs[j];
        s0 += w0[j] * b;
        s1 += w1[j] * b;
    }
    Wcs[idx * 2 + 0] = s0;
    Wcs[idx * 2 + 1] = s1;
}

// ---------------------------------------------------------------------------
// Kernel C: cbias = b2 @ basis^T + xstar   (512)
// ---------------------------------------------------------------------------
__global__ __launch_bounds__(256) void cbias_kernel(
    const float* __restrict__ b2, const float* __restrict__ basis,
    const float* __restrict__ xstar, float* __restrict__ cbias)
{
    int c = blockIdx.x * blockDim.x + threadIdx.x;
    if (c >= CDIM) return;
    const float* bs = basis + c * KFREE;
    float t = xstar[c];
    for (int j = 0; j < KFREE; ++j) t += b2[j] * bs[j];
    cbias[c] = t;
}

// ---------------------------------------------------------------------------
// Main kernel: leakyrelu(x@W1+b1) @ Wc + cbias  -> polynomial epilogue
// One wave32 per 16-row tile, 2 waves / block. fp32 WMMA 16x16x4.
// x tile staged memory->LDS with GLOBAL_LOAD_ASYNC_TO_LDS_B128 (ASYNCcnt).
// ---------------------------------------------------------------------------
__global__ __launch_bounds__(64) void smile_main(
    const float* __restrict__ x,   const float* __restrict__ b1,
    const float* __restrict__ W1s, const float* __restrict__ Wcs,
    const float* __restrict__ cbias, float* __restrict__ out)
{
    __shared__ __align__(16) float xbuf[2][16][D_IN];  // 16 KB: x tiles
    __shared__ __align__(16) float hbuf[2][16][D_H];   // 32 KB: hidden act
    __shared__ __align__(16) float sbuf[2][16][16];    //  2 KB: sol bounce

    const int wv    = threadIdx.x >> 5;
    const int lane  = threadIdx.x & 31;
    const int m     = lane & 15;          // row (A) / col (B,C)
    const int ks    = (lane >> 4) << 1;   // 0 or 2 (fp32 16x4 A K-split)
    const int rhalf = (lane >> 4) << 3;   // C-layout row offset 0 / 8
    const int r0    = (blockIdx.x * 2 + wv) * 16;

    // ---- Phase 0: stage x tile (16 x 128) into LDS ------------------------
    {
        const float4* xg = (const float4*)(x + (size_t)r0 * D_IN) + lane;
        float4* xb = (float4*)&xbuf[wv][0][0] + lane;
#if HAVE_ASYNC_LDS
        #pragma unroll
        for (int i = 0; i < 16; ++i)
            __builtin_amdgcn_global_load_async_to_lds_b128(
                (as1_v4i)(xg + i * 32), (as3_v4i)(xb + i * 32), 0, 0);
#if __has_builtin(__builtin_amdgcn_s_wait_asynccnt)
        __builtin_amdgcn_s_wait_asynccnt(0);
        asm volatile("" ::: "memory");
#else
        asm volatile("s_wait_asynccnt 0" ::: "memory");
#endif
#else
        #pragma unroll
        for (int i = 0; i < 16; ++i)
            xb[i * 32] = xg[i * 32];
#endif
    }

    // ---- Phase 1: h = leakyrelu(x @ W1 + b1) into LDS ---------------------
    for (int nt = 0; nt < 16; ++nt) {
        v8f acc = {0.f,0.f,0.f,0.f,0.f,0.f,0.f,0.f};
        const float* w1p = W1s + ((nt << 10) + lane) * 2;   // + k*64 per step
        #pragma unroll 8
        for (int k = 0; k < 32; ++k) {
            const int kr = 4 * k + ks;
            v2f af = *(const v2f*)&xbuf[wv][m][kr];         // ds_load_b64
            v2f bf = *(const v2f*)(w1p + (k << 6));         // global_load_b64
            acc = __builtin_amdgcn_wmma_f32_16x16x4_f32(
                false, af, false, bf, (short)0, acc, false, false);
        }
        const float bias = b1[nt * 16 + m];
        #pragma unroll
        for (int i = 0; i < 8; ++i) {
            float v = acc[i] + bias;
            v = (v > 0.f) ? v : 0.01f * v;                  // leaky_relu(0.01)
            hbuf[wv][i + rhalf][nt * 16 + m] = v;
        }
    }

    // ---- Phase 2: sol tile = h @ Wc + cbias, then polynomial epilogue -----
    for (int ct = 0; ct < 32; ++ct) {
        v8f acc = {0.f,0.f,0.f,0.f,0.f,0.f,0.f,0.f};
        const float* wcp = Wcs + ((ct << 11) + lane) * 2;   // + k*64 per step
        __builtin_prefetch(wcp + 4096 + (lane << 6), 0, 0); // next tile
        #pragma unroll 8
        for (int k = 0; k < 64; ++k) {
            const int kr = 4 * k + ks;
            v2f af = *(const v2f*)&hbuf[wv][m][kr];         // ds_load_b64
            v2f bf = *(const v2f*)(wcp + (k << 6));         // global_load_b64
            acc = __builtin_amdgcn_wmma_f32_16x16x4_f32(
                false, af, false, bf, (short)0, acc, false, false);
        }
        const float bias = cbias[ct * 16 + m];
        #pragma unroll
        for (int i = 0; i < 8; ++i)
            sbuf[wv][i + rhalf][m] = acc[i] + bias;

        // 64 (row, segment) tasks per 16-col tile; each lane takes 2
        #pragma unroll
        for (int tt = 0; tt < 2; ++tt) {
            const int t   = lane + (tt << 5);
            const int row = t >> 2;
            const int sq  = t & 3;
            const float4 s4 = *(const float4*)&sbuf[wv][row][sq << 2];
            // descending [x^3,x^2,x,1] -> ascending a0..a3
            const float a0 = s4.w, a1 = s4.z, a2 = s4.y, a3 = s4.x;

            float A2[7];
            A2[0] = a0*a0;
            A2[1] = 2.f*a0*a1;
            A2[2] = 2.f*a0*a2 + a1*a1;
            A2[3] = 2.f*(a0*a3 + a1*a2);
            A2[4] = 2.f*a1*a3 + a2*a2;
            A2[5] = 2.f*a2*a3;
            A2[6] = a3*a3;

            const float bb0 = a1, bb1 = 2.f*a2, bb2 = 3.f*a3;   // w'
            float B2[5];
            B2[0] = bb0*bb0;
            B2[1] = 2.f*bb0*bb1;
            B2[2] = 2.f*bb0*bb2 + bb1*bb1;
            B2[3] = 2.f*bb1*bb2;
            B2[4] = bb2*bb2;

            const float cc0 = 2.f*a2, cc1 = 6.f*a3;             // w''

            float cond[CONDL];
            #pragma unroll
            for (int j = 0; j < CONDL; ++j) cond[j] = 0.f;

            // -a^2 * b'^2 / 4      (len 11)
            #pragma unroll
            for (int i = 0; i < 7; ++i)
                #pragma unroll
                for (int j = 0; j < 5; ++j)
                    cond[i + j] -= 0.25f * A2[i] * B2[j];
            // +2 a^2 c  +4 a^2     (len 8 / 7)
            #pragma unroll
            for (int i = 0; i < 7; ++i) {
                cond[i]     += 2.f * A2[i] * cc0;
                cond[i + 1] += 2.f * A2[i] * cc1;
                cond[i]     += 4.f * A2[i];
            }
            // -a b'^2  -4 a b' x   (len 8 / 7)
            const float av[4] = {a0, a1, a2, a3};
            const float bv[3] = {bb0, bb1, bb2};
            #pragma unroll
            for (int i = 0; i < 4; ++i) {
                #pragma unroll
                for (int j = 0; j < 5; ++j) cond[i + j]     -= av[i] * B2[j];
                #pragma unroll
                for (int j = 0; j < 3; ++j) cond[i + j + 1] -= 4.f * av[i] * bv[j];
            }
            // + b'^2 x^2           (len 7)
            #pragma unroll
            for (int j = 0; j < 5; ++j) cond[j + 2] += B2[j];

            const int seg = (ct << 2) + sq;
            float* op = out + ((size_t)(r0 + row) * NSEG + seg) * CONDL;
            #pragma unroll
            for (int j = 0; j < CONDL; ++j) op[j] = cond[j];
        }
    }
}

// ---------------------------------------------------------------------------
extern "C" void kernel_launch(void* const* d_in, const int* in_sizes, int n_in,
                              void* d_out, int out_size, void* d_ws, size_t ws_size,
                              hipStream_t stream) {
    const float* x     = (const float*)d_in[0];
    const float* W1    = (const float*)d_in[1];
    const float* b1    = (const float*)d_in[2];
    const float* W2    = (const float*)d_in[3];
    const float* b2    = (const float*)d_in[4];
    const float* basis = (const float*)d_in[5];
    const float* xstar = (const float*)d_in[6];
    float* out   = (float*)d_out;

    float* Wcs   = (float*)d_ws;              // 512 KB
    float* W1s   = Wcs + WCS_FLOATS;          // 128 KB
    float* cbias = W1s + W1S_FLOATS;          //   2 KB

    swizzle_w1 <<<(16 * 32 * 32 + 255) / 256, 256, 0, stream>>>(W1, W1s);
    wc_swizzled<<<(32 * 64 * 32 + 255) / 256, 256, 0, stream>>>(W2, basis, Wcs);
    cbias_kernel<<<(CDIM + 255) / 256, 256, 0, stream>>>(b2, basis, xstar, cbias);
    smile_main <<<B_ROWS / 32, 64, 0, stream>>>(x, b1, W1s, Wcs, cbias, out);
}